// GAT_22213570855135
// MI455X (gfx1250) — compile-verified
//
#include <hip/hip_runtime.h>
#include <hip/hip_bf16.h>
#include <math.h>

#define N_NODES 50000
#define N_EDGES 800000
#define IN_DIM  128
#define HID     128
#define OUT_DIM 64
#define NEG_SLOPE 0.2f

typedef __attribute__((ext_vector_type(2))) float v2f;
typedef __attribute__((ext_vector_type(8))) float v8f;

// ---------------------------------------------------------------------------
// GEMM: H[M x N] = X[M x K] @ W[K x N]; K == 128, N in {128, 64}, M % 16 == 0.
// Block = 128 threads (4 waves). Each block: one 16-row M tile x 64-col panel.
// Wave w owns columns [16*w, 16*w+16) of the panel via V_WMMA_F32_16X16X4_F32.
// W panel (128x64 f32 = 32KB) staged in LDS once per block.
// ---------------------------------------------------------------------------
__global__ __launch_bounds__(128)
void gat_gemm_wmma(const float* __restrict__ X, const float* __restrict__ W,
                   float* __restrict__ H, int M, int K, int N) {
  __shared__ float sW[128 * 64];
  const int tid = threadIdx.x;
  const int n0block = blockIdx.y * 64;
  for (int i = tid; i < K * 64; i += 128) {
    int kr = i >> 6, c = i & 63;
    sW[i] = W[(size_t)kr * N + n0block + c];
  }
  __syncthreads();

  const int lane = tid & 31;
  const int wave = tid >> 5;
  const int m0   = blockIdx.x * 16;
  const int half = lane >> 4;       // 0: lanes 0-15 (K pair {0,1}), 1: lanes 16-31 ({2,3})
  const int l16  = lane & 15;
  const int row  = m0 + l16;        // A-frag M index
  const int koff = half * 2;
  const int col  = wave * 16 + l16; // column within 64-wide panel

  v8f acc = {0.f, 0.f, 0.f, 0.f, 0.f, 0.f, 0.f, 0.f};
  const float* xrow = X + (size_t)row * K;
  for (int k = 0; k < K; k += 4) {
    v2f a, b;
    float2 av = *(const float2*)(xrow + k + koff);   // contiguous K pair -> b64 load
    a.x = av.x; a.y = av.y;
    b.x = sW[(k + koff) * 64 + col];
    b.y = sW[(k + koff + 1) * 64 + col];
    // D = A(16x4 f32) * B(4x16 f32) + C ; 8 args: neg_a, A, neg_b, B, c_mod, C, reuse_a, reuse_b
    acc = __builtin_amdgcn_wmma_f32_16x16x4_f32(false, a, false, b, (short)0, acc,
                                                false, false);
  }
  // C/D layout: VGPR r holds M = m0 + 8*half + r, N = col
  float* hb = H + (size_t)(m0 + half * 8) * N + (n0block + col);
  #pragma unroll
  for (int r = 0; r < 8; ++r) hb[(size_t)r * N] = acc[r];
}

// one wave per node row: el = h.al, er = h.ar
__global__ __launch_bounds__(256)
void gat_rowdot(const float* __restrict__ h, const float* __restrict__ al,
                const float* __restrict__ ar, float* __restrict__ el,
                float* __restrict__ er, int n, int D) {
  int gw = (blockIdx.x * blockDim.x + threadIdx.x) >> 5;
  if (gw >= n) return;
  int lane = threadIdx.x & 31;
  const float* hr = h + (size_t)gw * D;
  float accl = 0.f, accr = 0.f;
  for (int c = lane; c < D; c += 32) {
    float v = hr[c];
    accl = fmaf(v, al[c], accl);
    accr = fmaf(v, ar[c], accr);
  }
  #pragma unroll
  for (int off = 16; off > 0; off >>= 1) {
    accl += __shfl_down(accl, off, 32);
    accr += __shfl_down(accr, off, 32);
  }
  if (lane == 0) { el[gw] = accl; er[gw] = accr; }
}

__global__ void gat_fill(float* __restrict__ p, float v, int n) {
  int i = blockIdx.x * blockDim.x + threadIdx.x;
  if (i < n) p[i] = v;
}

// float atomic max via monotone int/uint mapping (init must be -inf)
__device__ __forceinline__ void atomicMaxF32(float* addr, float val) {
  if (val >= 0.f) atomicMax((int*)addr, __float_as_int(val));
  else            atomicMin((unsigned int*)addr, __float_as_uint(val));
}

__global__ __launch_bounds__(256)
void gat_scores(const float* __restrict__ el, const float* __restrict__ er,
                const int* __restrict__ src, const int* __restrict__ dst,
                float* __restrict__ e, float* __restrict__ m, int E) {
  int i = blockIdx.x * blockDim.x + threadIdx.x;
  if (i >= E) return;
  float v = el[src[i]] + er[dst[i]];
  v = (v > 0.f) ? v : NEG_SLOPE * v;
  e[i] = v;
  atomicMaxF32(&m[dst[i]], v);
}

__global__ __launch_bounds__(256)
void gat_expsum(float* __restrict__ e, const int* __restrict__ dst,
                const float* __restrict__ m, float* __restrict__ s, int E) {
  int i = blockIdx.x * blockDim.x + threadIdx.x;
  if (i >= E) return;
  int d = dst[i];
  float ex = __expf(e[i] - m[d]);   // m[d] finite: node d has >=1 incoming edge
  e[i] = ex;                        // in-place e -> exp(e - m)
  atomicAdd(&s[d], ex);
}

// one wave per edge; lanes stripe the feature dim (float4 gather, f32 atomics scatter)
__global__ __launch_bounds__(256)
void gat_aggregate(const float* __restrict__ ex, const float* __restrict__ s,
                   const int* __restrict__ src, const int* __restrict__ dst,
                   const float* __restrict__ h, float* __restrict__ out,
                   int E, int D) {
  int w = (blockIdx.x * blockDim.x + threadIdx.x) >> 5;
  if (w >= E) return;
  int lane = threadIdx.x & 31;
  int sN = src[w], dN = dst[w];
  float alpha = ex[w] / s[dN];
  if (D == 128) {
    float4 hv = *(const float4*)(h + (size_t)sN * 128 + lane * 4);
    float* o = out + (size_t)dN * 128 + lane * 4;
    atomicAdd(o + 0, alpha * hv.x);
    atomicAdd(o + 1, alpha * hv.y);
    atomicAdd(o + 2, alpha * hv.z);
    atomicAdd(o + 3, alpha * hv.w);
  } else { // D == 64
    float2 hv = *(const float2*)(h + (size_t)sN * 64 + lane * 2);
    float* o = out + (size_t)dN * 64 + lane * 2;
    atomicAdd(o + 0, alpha * hv.x);
    atomicAdd(o + 1, alpha * hv.y);
  }
}

__global__ void gat_relu(float* __restrict__ p, unsigned int n) {
  unsigned int i = blockIdx.x * blockDim.x + threadIdx.x;
  if (i < n) p[i] = fmaxf(p[i], 0.f);
}

// column-sum readout: thread owns col = tid%64, strides rows, one atomic at end
__global__ __launch_bounds__(256)
void gat_readout(const float* __restrict__ h3, float* __restrict__ out,
                 int n, int D) {
  int col = threadIdx.x & 63;
  int rowStart = blockIdx.x * (blockDim.x >> 6) + (threadIdx.x >> 6);
  int rowStride = gridDim.x * (blockDim.x >> 6);
  float acc = 0.f;
  for (int r = rowStart; r < n; r += rowStride) acc += h3[(size_t)r * D + col];
  atomicAdd(&out[col], acc);
}

extern "C" void kernel_launch(void* const* d_in, const int* in_sizes, int n_in,
                              void* d_out, int out_size, void* d_ws, size_t ws_size,
                              hipStream_t stream) {
  const float* x   = (const float*)d_in[0];
  const int*   src = (const int*)  d_in[1];
  const int*   dst = (const int*)  d_in[2];
  const float* W1  = (const float*)d_in[3];
  const float* al1 = (const float*)d_in[4];
  const float* ar1 = (const float*)d_in[5];
  const float* W2  = (const float*)d_in[6];
  const float* al2 = (const float*)d_in[7];
  const float* ar2 = (const float*)d_in[8];
  const float* W3  = (const float*)d_in[9];
  const float* al3 = (const float*)d_in[10];
  const float* ar3 = (const float*)d_in[11];

  float* ws = (float*)d_ws;
  size_t off = 0;
  float* P  = ws + off; off += (size_t)N_NODES * HID;  // projected features h
  float* A  = ws + off; off += (size_t)N_NODES * HID;  // layer output ping
  float* B  = ws + off; off += (size_t)N_NODES * HID;  // layer output pong
  float* el = ws + off; off += N_NODES;
  float* er = ws + off; off += N_NODES;
  float* m  = ws + off; off += N_NODES;
  float* s  = ws + off; off += N_EDGES > N_NODES ? N_NODES : N_NODES;
  float* e  = ws + off; off += N_EDGES;

  auto layer = [&](const float* xin, const float* W, const float* al,
                   const float* ar, int Din, int Dout, float* hproj,
                   float* aggout) {
    dim3 gg(N_NODES / 16, Dout / 64, 1);
    gat_gemm_wmma<<<gg, 128, 0, stream>>>(xin, W, hproj, N_NODES, Din, Dout);
    gat_rowdot<<<(N_NODES * 32 + 255) / 256, 256, 0, stream>>>(hproj, al, ar, el,
                                                               er, N_NODES, Dout);
    gat_fill<<<(N_NODES + 255) / 256, 256, 0, stream>>>(m, -INFINITY, N_NODES);
    hipMemsetAsync(s, 0, N_NODES * sizeof(float), stream);
    hipMemsetAsync(aggout, 0, (size_t)N_NODES * Dout * sizeof(float), stream);
    gat_scores<<<(N_EDGES + 255) / 256, 256, 0, stream>>>(el, er, src, dst, e, m,
                                                          N_EDGES);
    gat_expsum<<<(N_EDGES + 255) / 256, 256, 0, stream>>>(e, dst, m, s, N_EDGES);
    gat_aggregate<<<(N_EDGES * 32 + 255) / 256, 256, 0, stream>>>(e, s, src, dst,
                                                                  hproj, aggout,
                                                                  N_EDGES, Dout);
    unsigned int n = (unsigned int)N_NODES * Dout;
    gat_relu<<<(n + 255) / 256, 256, 0, stream>>>(aggout, n);
  };

  layer(x, W1, al1, ar1, IN_DIM, HID, P, A);     // h1 -> A
  layer(A, W2, al2, ar2, HID, HID, P, B);        // h2 -> B
  layer(B, W3, al3, ar3, HID, OUT_DIM, P, A);    // h3 -> A (first N*64 floats)

  hipMemsetAsync(d_out, 0, OUT_DIM * sizeof(float), stream);
  gat_readout<<<512, 256, 0, stream>>>(A, (float*)d_out, N_NODES, OUT_DIM);
}